// SoftSkeletonize_35579509080158
// MI455X (gfx1250) — compile-verified
//
#include <hip/hip_runtime.h>
#include <stdint.h>

// Soft-skeletonize, fused: per step, one kernel computes E = erode(X) (7-pt star min),
// open = dilate(E) (27-pt max), delta = relu(X - open), skel += relu(delta - skel*delta),
// and writes X' = E. LDS-tiled with halo 2. Interior tiles staged via TDM
// tensor_load_to_lds; boundary tiles staged via per-lane async global->LDS copies.

#define TD   8            // output tile edge per block
#define XD  (TD + 4)      // 12: input tile with halo 2
#define ED  (TD + 2)      // 10: eroded tile with halo 1
#define NT  (TD * TD * TD)
#define DIM  128
#define DIM2 (DIM * DIM)
#define DIM3 (DIM * DIM * DIM)

#define LXI(x, y, z) ((z) * (XD * XD) + (y) * XD + (x))
#define LEI(x, y, z) ((z) * (ED * ED) + (y) * ED + (x))

#if defined(__gfx1250__) && __has_builtin(__builtin_amdgcn_tensor_load_to_lds) && \
    __has_builtin(__builtin_amdgcn_s_wait_tensorcnt)
#define HAVE_TDM 1
#else
#define HAVE_TDM 0
#endif

#if defined(__gfx1250__) && __has_builtin(__builtin_amdgcn_global_load_async_to_lds_b32) && \
    __has_builtin(__builtin_amdgcn_s_wait_asynccnt)
#define HAVE_ASYNC 1
#else
#define HAVE_ASYNC 0
#endif

typedef uint32_t u32x4 __attribute__((ext_vector_type(4)));
typedef int      i32x8 __attribute__((ext_vector_type(8)));
typedef int      i32x4 __attribute__((ext_vector_type(4)));

__global__ __launch_bounds__(NT)
void skel_step(const float* __restrict__ X, const float* __restrict__ skel_in,
               float* __restrict__ skel_out, float* __restrict__ Xout, int init)
{
    __shared__ float lx[XD * XD * XD];   // input tile, halo 2 (+inf padded)
    __shared__ float le[ED * ED * ED];   // eroded tile, halo 1 (-inf at global OOB)

    const int tx = threadIdx.x, ty = threadIdx.y, tz = threadIdx.z;
    const int tid = tz * (TD * TD) + ty * TD + tx;
    const int ox = blockIdx.x * TD;
    const int oy = blockIdx.y * TD;
    const int vol = blockIdx.z >> 4;            // 16 tiles per axis
    const int oz = (blockIdx.z & 15) * TD;
    const float* __restrict__ Xv = X + (size_t)vol * DIM3;

    const size_t gidx = (size_t)vol * DIM3 +
                        (size_t)(oz + tz) * DIM2 + (size_t)(oy + ty) * DIM + (size_t)(ox + tx);
    if (!init)
        __builtin_prefetch(skel_in + gidx, 0, 0);   // global_prefetch_b8: hide skel read latency

    // ---------------- stage input tile into LDS ----------------
    const bool interior =
        (ox >= 2) && (ox <= DIM - XD + 2) &&
        (oy >= 2) && (oy <= DIM - XD + 2) &&
        (oz >= 2) && (oz <= DIM - XD + 2);
    (void)interior;
#if HAVE_TDM
    if (interior) {
        if (tid < 32) {   // one wave issues the TDM descriptor (scalar op)
            const uint64_t ga = (uint64_t)(uintptr_t)(Xv +
                (size_t)(oz - 2) * DIM2 + (size_t)(oy - 2) * DIM + (size_t)(ox - 2));
            const uint32_t lds_off = (uint32_t)(uintptr_t)&lx[0]; // low 32b of generic ptr = LDS offset
            // D# group 0: count=1 | lds_addr | global_addr[56:0] | type=2
            u32x4 g0 = { 1u,
                         lds_off,
                         (uint32_t)ga,
                         (uint32_t)(ga >> 32) | (2u << 30) };
            // D# group 1: data_size=4B; tensor_dim0/1=128; tile 12x12x12; strides 128, 16384
            i32x8 g1 = { (int)(2u << 16),                     // data_size = 2 (4 bytes)
                         (int)((uint32_t)DIM  << 16),         // tensor_dim0 lo16 (=128)
                         (int)((uint32_t)DIM  << 16),         // dim0 hi16=0 | tensor_dim1 lo16
                         (int)((uint32_t)XD   << 16),         // dim1 hi16=0 | tile_dim0 = 12
                         (int)(((uint32_t)XD << 16) | XD),    // tile_dim1=12 | tile_dim2=12
                         DIM,                                 // tensor_dim0_stride = 128
                         (int)((uint32_t)DIM2 << 16),         // stride0 hi=0 | stride1 lo16 = 16384
                         0 };                                 // stride1 hi = 0
            // D# group 2: tensor_dim2=128; tensor_dim3=1; tensor_dim2_stride=128^3; tile_dim3=0
            i32x4 g2 = { DIM, 1, DIM3, 0 };
            i32x4 g3 = { 0, 0, 0, 0 };
            i32x8 g4 = { 0, 0, 0, 0, 0, 0, 0, 0 };  // extra group (clang-23 6-arg form), unused
            __builtin_amdgcn_tensor_load_to_lds(g0, g1, g2, g3, g4, 0);
            __builtin_amdgcn_s_wait_tensorcnt(0);
        }
    } else
#endif
    {
#if HAVE_ASYNC
        // min-pool identity pre-fill, then async-copy the in-range intersection
        for (int i = tid; i < XD * XD * XD; i += NT)
            lx[i] = __builtin_inff();
        __syncthreads();
        for (int i = tid; i < XD * XD * XD; i += NT) {
            const int x = i % XD;
            const int r = i / XD;
            const int y = r % XD;
            const int z = r / XD;
            const int gx = ox - 2 + x, gy = oy - 2 + y, gz = oz - 2 + z;
            if ((unsigned)gx < DIM && (unsigned)gy < DIM && (unsigned)gz < DIM)
                __builtin_amdgcn_global_load_async_to_lds_b32(
                    (int*)(Xv + ((size_t)gz * DIM2 + (size_t)gy * DIM + (size_t)gx)),
                    (int*)&lx[i], 0, 0);
        }
        __builtin_amdgcn_s_wait_asynccnt(0);
#else
        for (int i = tid; i < XD * XD * XD; i += NT) {
            const int x = i % XD;
            const int r = i / XD;
            const int y = r % XD;
            const int z = r / XD;
            const int gx = ox - 2 + x, gy = oy - 2 + y, gz = oz - 2 + z;
            float v = __builtin_inff();
            if ((unsigned)gx < DIM && (unsigned)gy < DIM && (unsigned)gz < DIM)
                v = Xv[(size_t)gz * DIM2 + (size_t)gy * DIM + (size_t)gx];
            lx[i] = v;
        }
#endif
    }
    __syncthreads();

    // ---------------- E = erode(X): 7-point star min over (TD+2)^3 ----------------
    for (int i = tid; i < ED * ED * ED; i += NT) {
        const int x = i % ED;
        const int r = i / ED;
        const int y = r % ED;
        const int z = r / ED;
        const int cx = x + 1, cy = y + 1, cz = z + 1;
        float m = lx[LXI(cx, cy, cz)];
        m = fminf(m, lx[LXI(cx - 1, cy, cz)]);
        m = fminf(m, lx[LXI(cx + 1, cy, cz)]);
        m = fminf(m, lx[LXI(cx, cy - 1, cz)]);
        m = fminf(m, lx[LXI(cx, cy + 1, cz)]);
        m = fminf(m, lx[LXI(cx, cy, cz - 1)]);
        m = fminf(m, lx[LXI(cx, cy, cz + 1)]);
        // positions outside the global tensor must act as -inf for the following max-pool
        const int gx = ox - 1 + x, gy = oy - 1 + y, gz = oz - 1 + z;
        if (!((unsigned)gx < DIM && (unsigned)gy < DIM && (unsigned)gz < DIM))
            m = -__builtin_inff();
        le[i] = m;
    }
    __syncthreads();

    // ---------------- open = dilate(E); pointwise skeleton update ----------------
    float mx = -__builtin_inff();
#pragma unroll
    for (int dz = 0; dz < 3; ++dz)
#pragma unroll
        for (int dy = 0; dy < 3; ++dy)
#pragma unroll
            for (int dx = 0; dx < 3; ++dx)
                mx = fmaxf(mx, le[LEI(tx + dx, ty + dy, tz + dz)]);

    const float Xc = lx[LXI(tx + 2, ty + 2, tz + 2)];
    const float Ec = le[LEI(tx + 1, ty + 1, tz + 1)];
    const float delta = fmaxf(0.0f, Xc - mx);

    float s;
    if (init) {
        s = delta;                                  // skel = relu(img - open(img))
    } else {
        const float s0 = skel_in[gidx];
        s = s0 + fmaxf(0.0f, __builtin_fmaf(-s0, delta, delta)); // skel += relu(delta - skel*delta)
    }
    skel_out[gidx] = s;
    Xout[gidx] = Ec;                                // next image = erode(X)
}

extern "C" void kernel_launch(void* const* d_in, const int* in_sizes, int n_in,
                              void* d_out, int out_size, void* d_ws, size_t ws_size,
                              hipStream_t stream)
{
    (void)n_in; (void)out_size; (void)ws_size;
    const float* img = (const float*)d_in[0];
    float* skel = (float*)d_out;
    const int N = in_sizes[0];            // 2*2*128^3 = 8,388,608
    const int vols = N / DIM3;            // 4 independent volumes

    float* A = (float*)d_ws;              // image ping
    float* B = A + N;                     // image pong (needs 2*N*4 = 64 MB scratch)

    dim3 grid(DIM / TD, DIM / TD, (DIM / TD) * vols);
    dim3 block(TD, TD, TD);

    // step 0: skel = relu(img - open(img)); A = erode(img)
    skel_step<<<grid, block, 0, stream>>>(img, skel, skel, A, 1);

    const float* cur = A;
    float* nxt = B;
    for (int it = 0; it < 40; ++it) {
        skel_step<<<grid, block, 0, stream>>>(cur, skel, skel, nxt, 0);
        float* t = (float*)cur; cur = nxt; nxt = t;
    }
}